// MultiHeadSelfAttention_14783277433120
// MI455X (gfx1250) — compile-verified
//
#include <hip/hip_runtime.h>
#include <hip/hip_bf16.h>

// MHA forward, fp32, MI455X (gfx1250, wave32).
// B=2 S=2048 E=768 H=12 D=64.  Bandwidth-bound on the 402MB attention-weights
// write => keep fp32 precision, use V_WMMA_F32_16X16X4_F32, and fuse
// scores+softmax+PV so the big tensor touches HBM exactly once.

typedef __attribute__((ext_vector_type(2))) float v2f;
typedef __attribute__((ext_vector_type(8))) float v8f;

static constexpr int Bc = 2;
static constexpr int Sc = 2048;
static constexpr int Ec = 768;
static constexpr int Hc = 12;
static constexpr int Dc = 64;
static constexpr int Mtok = Bc * Sc;           // 4096 token rows

__device__ __forceinline__ v8f wmma_f32_4(v2f a, v2f b, v8f c) {
  // 8 args: (neg_a, A, neg_b, B, c_mod, C, reuse_a, reuse_b)
  return __builtin_amdgcn_wmma_f32_16x16x4_f32(false, a, false, b, (short)0, c,
                                               false, false);
}

// ---------------------------------------------------------------------------
// Kernel 1: fused QKV projection.  One wave32 per 16x16 output tile.
//   y = x @ W + b ; Q additionally scaled by 1/sqrt(D)=0.125.
//   Output layout: (B*H, S, D) row-major per head.
// A-frag (16x4 f32): lane<16 -> {K=k,k+1}, lane>=16 -> {K=k+2,k+3}, M=lane%16.
// B-frag (4x16 f32): lane<16 -> rows k,k+1 at N=lane, lane>=16 -> rows k+2,k+3.
// ---------------------------------------------------------------------------
__global__ __launch_bounds__(32) void qkv_proj_kernel(
    const float* __restrict__ x,
    const float* __restrict__ Wq, const float* __restrict__ bq,
    const float* __restrict__ Wk, const float* __restrict__ bk,
    const float* __restrict__ Wv, const float* __restrict__ bv,
    float* __restrict__ Q, float* __restrict__ K, float* __restrict__ V) {
  const int lane = threadIdx.x & 31;
  const int hf   = lane >> 4;        // which K-pair this lane holds
  const int l16  = lane & 15;
  const int m0   = blockIdx.x * 16;  // token-row tile
  const int n0   = blockIdx.y * 16;  // output-feature tile
  const int which = blockIdx.z;      // 0=Q 1=K 2=V

  const float* W    = (which == 0) ? Wq : (which == 1) ? Wk : Wv;
  const float* bias = (which == 0) ? bq : (which == 1) ? bk : bv;
  float*       dst  = (which == 0) ? Q  : (which == 1) ? K  : V;

  const float* arow = x + (size_t)(m0 + l16) * Ec;

  v8f acc = {};
  for (int k = 0; k < Ec; k += 4) {
    v2f a = *(const v2f*)(arow + k + 2 * hf);          // contiguous pair
    v2f bf;
    bf.x = W[(size_t)(k + 2 * hf) * Ec + n0 + l16];
    bf.y = W[(size_t)(k + 2 * hf + 1) * Ec + n0 + l16];
    acc = wmma_f32_4(a, bf, acc);
  }

  const float bia   = bias[n0 + l16];                  // col depends only on l16
  const float scale = (which == 0) ? 0.125f : 1.0f;    // fold 1/sqrt(D) into Q

#pragma unroll
  for (int r = 0; r < 8; ++r) {
    const int row = m0 + r + hf * 8;                   // D-matrix layout
    const int col = n0 + l16;
    const int b = row >> 11, s = row & (Sc - 1);
    const int h = col >> 6,  d = col & (Dc - 1);
    dst[((size_t)(b * Hc + h) * Sc + s) * Dc + d] = (acc[r] + bia) * scale;
  }
}

// ---------------------------------------------------------------------------
// Kernel 2: fused attention for one (b,h) x 16-row q-block.
//   256 threads = 8 waves; LDS holds the full 16x2048 score block (128KB).
//   Phase 1: scores = Q Kt  (wave w owns 16 column tiles)
//   Phase 2: row softmax in LDS, normalized P streamed once to d_out
//   Phase 3: O = P @ V, K-dim split across wave pairs, partials via LDS
// ---------------------------------------------------------------------------
__global__ __launch_bounds__(256) void attn_fused_kernel(
    const float* __restrict__ Q, const float* __restrict__ Kmat,
    const float* __restrict__ V, float* __restrict__ Pout,
    float* __restrict__ AO) {
  extern __shared__ float smem[];
  float* lds_p = smem;                  // 16 * 2048 floats
  float* lds_o = smem + 16 * Sc;        // 4 * 256 floats (PV partials)

  const int tid  = threadIdx.x;
  const int wave = tid >> 5;
  const int lane = tid & 31;
  const int hf   = lane >> 4;
  const int l16  = lane & 15;
  const int q0   = blockIdx.x * 16;
  const int bh   = blockIdx.y;
  const int b    = bh / Hc, h = bh % Hc;

  const float* Qb = Q    + ((size_t)bh * Sc + q0) * Dc;
  const float* Kb = Kmat +  (size_t)bh * Sc * Dc;
  const float* Vb = V    +  (size_t)bh * Sc * Dc;

  // ---- Phase 1: S = Q Kt for 16 column-tiles per wave -------------------
  // Q fragments depend only on k: preload all 16 (Q already carries 1/8).
  v2f qf[16];
#pragma unroll
  for (int k4 = 0; k4 < 16; ++k4)
    qf[k4] = *(const v2f*)(Qb + l16 * Dc + k4 * 4 + 2 * hf);

  for (int t = 0; t < 16; ++t) {
    const int nt = wave * 16 + t;                       // column tile index
    const float* Kt = Kb + (size_t)(nt * 16) * Dc;
    v8f acc = {};
#pragma unroll
    for (int k4 = 0; k4 < 16; ++k4) {
      // B = Kt^T: B[k][n] = K[n][k] -> contiguous pair along d
      v2f bf = *(const v2f*)(Kt + l16 * Dc + k4 * 4 + 2 * hf);
      acc = wmma_f32_4(qf[k4], bf, acc);
    }
#pragma unroll
    for (int r = 0; r < 8; ++r)
      lds_p[(r + hf * 8) * Sc + nt * 16 + l16] = acc[r];
  }
  __syncthreads();

  // ---- Phase 2: softmax per row (wave handles 2 rows) -------------------
  float* Pg = Pout + ((size_t)bh * Sc + q0) * Sc;
#pragma unroll
  for (int rr = 0; rr < 2; ++rr) {
    const int row = wave * 2 + rr;
    float* prow = lds_p + (size_t)row * Sc;
    float mx = -3.402823466e38f;
    for (int i = lane; i < Sc; i += 32) mx = fmaxf(mx, prow[i]);
#pragma unroll
    for (int off = 16; off; off >>= 1) mx = fmaxf(mx, __shfl_xor(mx, off, 32));
    float sum = 0.f;
    for (int i = lane; i < Sc; i += 32) {
      float e = __expf(prow[i] - mx);
      prow[i] = e;
      sum += e;
    }
#pragma unroll
    for (int off = 16; off; off >>= 1) sum += __shfl_xor(sum, off, 32);
    const float inv = 1.0f / sum;
    float* pg = Pg + (size_t)row * Sc;
    for (int i = lane; i < Sc; i += 32) {
      const float p = prow[i] * inv;
      prow[i] = p;                                     // keep for phase 3
      pg[i]   = p;                                     // the 402MB output, once
    }
  }
  __syncthreads();

  // ---- Phase 3: O = P @ V (16x2048 @ 2048x64) ---------------------------
  const int nt = wave & 3;            // which 16-col tile of D=64
  const int kh = wave >> 2;           // K-dim half (0..1023 / 1024..2047)
  v8f acc = {};
  for (int k = kh * 1024; k < kh * 1024 + 1024; k += 4) {
    v2f a = *(const v2f*)(lds_p + (size_t)l16 * Sc + k + 2 * hf);
    v2f bf;
    bf.x = Vb[(size_t)(k + 2 * hf) * Dc + nt * 16 + l16];
    bf.y = Vb[(size_t)(k + 2 * hf + 1) * Dc + nt * 16 + l16];
    acc = wmma_f32_4(a, bf, acc);
  }
  if (kh == 1) {
#pragma unroll
    for (int r = 0; r < 8; ++r)
      lds_o[nt * 256 + (r + hf * 8) * 16 + l16] = acc[r];
  }
  __syncthreads();
  if (kh == 0) {
#pragma unroll
    for (int r = 0; r < 8; ++r) {
      const float v = acc[r] + lds_o[nt * 256 + (r + hf * 8) * 16 + l16];
      const int srow = q0 + r + hf * 8;
      // AO layout: (B, S, E) so the final GEMM reads it row-major
      AO[((size_t)(b * Sc + srow)) * Ec + h * Dc + nt * 16 + l16] = v;
    }
  }
}

// ---------------------------------------------------------------------------
// Kernel 3: output projection  out = AO @ Wo + bo  -> d_out[0 : B*S*E)
// ---------------------------------------------------------------------------
__global__ __launch_bounds__(32) void out_proj_kernel(
    const float* __restrict__ A, const float* __restrict__ Wo,
    const float* __restrict__ bo, float* __restrict__ out) {
  const int lane = threadIdx.x & 31;
  const int hf   = lane >> 4;
  const int l16  = lane & 15;
  const int m0   = blockIdx.x * 16;
  const int n0   = blockIdx.y * 16;

  const float* arow = A + (size_t)(m0 + l16) * Ec;
  v8f acc = {};
  for (int k = 0; k < Ec; k += 4) {
    v2f a = *(const v2f*)(arow + k + 2 * hf);
    v2f bf;
    bf.x = Wo[(size_t)(k + 2 * hf) * Ec + n0 + l16];
    bf.y = Wo[(size_t)(k + 2 * hf + 1) * Ec + n0 + l16];
    acc = wmma_f32_4(a, bf, acc);
  }
  const float bia = bo[n0 + l16];
#pragma unroll
  for (int r = 0; r < 8; ++r) {
    const int row = m0 + r + hf * 8;
    const int col = n0 + l16;
    out[(size_t)row * Ec + col] = acc[r] + bia;
  }
}

extern "C" void kernel_launch(void* const* d_in, const int* in_sizes, int n_in,
                              void* d_out, int out_size, void* d_ws,
                              size_t ws_size, hipStream_t stream) {
  const float* x  = (const float*)d_in[0];
  const float* Wq = (const float*)d_in[1];
  const float* bq = (const float*)d_in[2];
  const float* Wk = (const float*)d_in[3];
  const float* bk = (const float*)d_in[4];
  const float* Wv = (const float*)d_in[5];
  const float* bv = (const float*)d_in[6];
  const float* Wo = (const float*)d_in[7];
  const float* bo = (const float*)d_in[8];

  float* out     = (float*)d_out;                 // (B,S,E)
  float* weights = out + (size_t)Bc * Sc * Ec;    // (B,H,S,S)

  // Workspace: Q, K, V (each (B*H,S,D)) + AO ((B,S,E)) = 48 MB total.
  float* Qf = (float*)d_ws;
  float* Kf = Qf + (size_t)Mtok * Ec;
  float* Vf = Kf + (size_t)Mtok * Ec;
  float* AO = Vf + (size_t)Mtok * Ec;

  // 1) QKV projections: 256 x 48 tiles x {Q,K,V}
  dim3 g1(Mtok / 16, Ec / 16, 3);
  qkv_proj_kernel<<<g1, 32, 0, stream>>>(x, Wq, bq, Wk, bk, Wv, bv, Qf, Kf, Vf);

  // 2) Fused scores+softmax+PV: 128 q-blocks x 24 (b,h); 132KB dynamic LDS
  dim3 g2(Sc / 16, Bc * Hc);
  size_t lds_bytes = (16 * Sc + 4 * 256) * sizeof(float);
  attn_fused_kernel<<<g2, 256, lds_bytes, stream>>>(Qf, Kf, Vf, weights, AO);

  // 3) Output projection into d_out
  dim3 g3(Mtok / 16, Ec / 16);
  out_proj_kernel<<<g3, 32, 0, stream>>>(AO, Wo, bo, out);
}